// Net_32229434589583
// MI455X (gfx1250) — compile-verified
//
#include <hip/hip_runtime.h>

#define N_NODES 10000
#define N_EDGES 320000
#define D_IN 128
#define D_H 256

typedef __attribute__((ext_vector_type(2))) float v2f;
typedef __attribute__((ext_vector_type(8))) float v8f;

__device__ __forceinline__ float lrelu(float x) { return x >= 0.0f ? x : 0.01f * x; }

__global__ void zero_kernel(float* __restrict__ p, int n) {
  int i = blockIdx.x * blockDim.x + threadIdx.x;
  if (i < n) p[i] = 0.0f;
}

__global__ void deg_kernel(const int* __restrict__ dst, float* __restrict__ deg, int E) {
  int e = blockIdx.x * blockDim.x + threadIdx.x;
  if (e < E) atomicAdd(&deg[dst[e]], 1.0f);
}

// acc[dst[e], :] += X[e, :]   (d = feature width, dq = d/4)
__global__ void scatter_edge_kernel(const float4* __restrict__ X, const int* __restrict__ dst,
                                    float* __restrict__ acc, int E, int dq, int d) {
  int i = blockIdx.x * blockDim.x + threadIdx.x;
  if (i >= E * dq) return;
  int e = i / dq, c = i - e * dq;
  float4 v = X[i];
  float* a = acc + (long)dst[e] * d + c * 4;
  atomicAdd(a + 0, v.x); atomicAdd(a + 1, v.y);
  atomicAdd(a + 2, v.z); atomicAdd(a + 3, v.w);
}

// acc[dst[e], :] += nin[src[e], :]
__global__ void gather_scatter_kernel(const float* __restrict__ nin, const int* __restrict__ src,
                                      const int* __restrict__ dst, float* __restrict__ acc,
                                      int E, int dq, int d) {
  int i = blockIdx.x * blockDim.x + threadIdx.x;
  if (i >= E * dq) return;
  int e = i / dq, c = i - e * dq;
  float4 v = ((const float4*)(nin + (long)src[e] * d))[c];
  float* a = acc + (long)dst[e] * d + c * 4;
  atomicAdd(a + 0, v.x); atomicAdd(a + 1, v.y);
  atomicAdd(a + 2, v.z); atomicAdd(a + 3, v.w);
}

__global__ void div_deg_kernel(float* __restrict__ node, const float* __restrict__ deg,
                               int N, int d) {
  int i = blockIdx.x * blockDim.x + threadIdx.x;
  if (i >= N * d) return;
  node[i] /= fmaxf(deg[i / d], 1.0f);
}

// Fused: edge value = lrelu(0.5*(g[src]+g[dst]) + bias); scatter-add into acc[dst] (d=256)
__global__ void edge_fuse_kernel(const float* __restrict__ g, const float* __restrict__ bias,
                                 const int* __restrict__ src, const int* __restrict__ dst,
                                 float* __restrict__ acc, int E) {
  int i = blockIdx.x * blockDim.x + threadIdx.x;
  if (i >= E * 64) return;
  int e = i >> 6, c = i & 63;
  int s = src[e], t = dst[e];
  float4 a = ((const float4*)(g + (long)s * 256))[c];
  float4 b = ((const float4*)(g + (long)t * 256))[c];
  float4 bb = ((const float4*)bias)[c];
  float4 v;
  v.x = lrelu(0.5f * (a.x + b.x) + bb.x);
  v.y = lrelu(0.5f * (a.y + b.y) + bb.y);
  v.z = lrelu(0.5f * (a.z + b.z) + bb.z);
  v.w = lrelu(0.5f * (a.w + b.w) + bb.w);
  float* o = acc + (long)t * 256 + c * 4;
  atomicAdd(o + 0, v.x); atomicAdd(o + 1, v.y);
  atomicAdd(o + 2, v.z); atomicAdd(o + 3, v.w);
}

// Fused final: msum[f] += sum_e lrelu(0.5*(g2[src]+g2[dst]) + bias)[f]
__global__ void edge_reduce_kernel(const float* __restrict__ g, const float* __restrict__ bias,
                                   const int* __restrict__ src, const int* __restrict__ dst,
                                   float* __restrict__ msum, int E) {
  const int EPB = 256;
  int f = threadIdx.x;                       // 256 threads: one feature each
  int e0 = blockIdx.x * EPB;
  int e1 = e0 + EPB; if (e1 > E) e1 = E;
  float b = bias[f];
  float acc = 0.0f;
  for (int e = e0; e < e1; ++e) {
    int s = src[e], t = dst[e];
    acc += lrelu(0.5f * (g[(long)s * 256 + f] + g[(long)t * 256 + f]) + b);
  }
  atomicAdd(&msum[f], acc);
}

// C[M, dout] = A[M, din] @ W[dout, din]^T using V_WMMA_F32_16X16X4_F32.
// One wave per 16x16 output tile; 8 waves (256 threads) per block.
__global__ void __launch_bounds__(256)
wmma_gemm_nt_kernel(const float* __restrict__ A, const float* __restrict__ W,
                    float* __restrict__ C, int M, int din, int dout) {
  int ntN = dout >> 4;
  int t = blockIdx.x * 8 + threadIdx.y;
  int tm = t / ntN, tn = t - tm * ntN;
  int m0 = tm * 16, n0 = tn * 16;
  if (m0 >= M) return;                       // wave-uniform; EXEC stays all-1
  int lane = threadIdx.x;
  int r = lane & 15;                         // M row (A) / N col (B,C)
  int h2 = (lane >> 4) << 1;                 // K offset selector per lane half
  const float* arow = A + (long)(m0 + r) * din;
  const float* wrow = W + (long)(n0 + r) * din;
  v8f acc = {0.f, 0.f, 0.f, 0.f, 0.f, 0.f, 0.f, 0.f};
  for (int k0 = 0; k0 < din; k0 += 4) {
    int kk = k0 + h2;
    v2f a; a.x = arow[kk]; a.y = arow[kk + 1];     // A 16x4 frag: v0=K(kk), v1=K(kk+1)
    v2f b; b.x = wrow[kk]; b.y = wrow[kk + 1];     // B 4x16 frag: B[k][n] = W[n][k]
    acc = __builtin_amdgcn_wmma_f32_16x16x4_f32(
        false, a, false, b, (short)0, acc, false, false);
  }
  int rbase = m0 + (lane >> 4) * 8;          // C: VGPR vr -> row vr (+8 for hi half)
#pragma unroll
  for (int vr = 0; vr < 8; ++vr)
    C[(long)(rbase + vr) * dout + n0 + r] = acc[vr];
}

__global__ void final_kernel(const float* __restrict__ msum, const float* __restrict__ W3,
                             const float* __restrict__ b3, float* __restrict__ out) {
  int o = threadIdx.x;                        // 64 threads
  const float invE = 1.0f / (float)N_EDGES;
  float s = b3[o];
  for (int j = 0; j < 2 * D_H; ++j)
    s += (msum[j] * invE) * W3[o * (2 * D_H) + j];
  out[o] = s;
}

extern "C" void kernel_launch(void* const* d_in, const int* in_sizes, int n_in,
                              void* d_out, int out_size, void* d_ws, size_t ws_size,
                              hipStream_t stream) {
  (void)in_sizes; (void)n_in; (void)out_size; (void)ws_size;
  // Workspace layout (fp32): 3 node buffers [10000,256], deg [10000], msum [512]
  float* buf0 = (float*)d_ws;
  float* buf1 = buf0 + (long)N_NODES * D_H;
  float* buf2 = buf1 + (long)N_NODES * D_H;
  float* deg  = buf2 + (long)N_NODES * D_H;
  float* msum = deg + N_NODES;

  const int TB = 256;
  auto cdiv = [](int a, int b) { return (a + b - 1) / b; };
  const int gemm_blocks = (N_NODES / 16) * (D_H / 16) / 8;   // 10000 tiles / 8 waves

  zero_kernel<<<cdiv(2 * D_H, TB), TB, 0, stream>>>(msum, 2 * D_H);

  for (int br = 0; br < 2; ++br) {
    const float* X  = (const float*)d_in[br * 3 + 0];
    const int* src  = (const int*)d_in[br * 3 + 1];
    const int* dst  = (const int*)d_in[br * 3 + 2];
    const float* Wa = (const float*)d_in[6 + br * 4 + 0];   // W11 / W21
    const float* ba = (const float*)d_in[6 + br * 4 + 1];   // b11 / b21
    const float* Wb = (const float*)d_in[6 + br * 4 + 2];   // W12 / W22
    const float* bb = (const float*)d_in[6 + br * 4 + 3];   // b12 / b22

    // degree (shared by both layers; same dst)
    zero_kernel<<<cdiv(N_NODES, TB), TB, 0, stream>>>(deg, N_NODES);
    deg_kernel<<<cdiv(N_EDGES, TB), TB, 0, stream>>>(dst, deg, N_EDGES);

    // ---- layer 1 ----
    zero_kernel<<<cdiv(N_NODES * D_IN, TB), TB, 0, stream>>>(buf0, N_NODES * D_IN);
    scatter_edge_kernel<<<cdiv(N_EDGES * (D_IN / 4), TB), TB, 0, stream>>>(
        (const float4*)X, dst, buf0, N_EDGES, D_IN / 4, D_IN);
    div_deg_kernel<<<cdiv(N_NODES * D_IN, TB), TB, 0, stream>>>(buf0, deg, N_NODES, D_IN);
    zero_kernel<<<cdiv(N_NODES * D_IN, TB), TB, 0, stream>>>(buf1, N_NODES * D_IN);
    gather_scatter_kernel<<<cdiv(N_EDGES * (D_IN / 4), TB), TB, 0, stream>>>(
        buf0, src, dst, buf1, N_EDGES, D_IN / 4, D_IN);
    // g1 = h2 @ Wa^T  (node-level GEMM, fp32 WMMA)
    wmma_gemm_nt_kernel<<<gemm_blocks, dim3(32, 8), 0, stream>>>(
        buf1, Wa, buf2, N_NODES, D_IN, D_H);

    // ---- layer 2 (edge relu fused into aggregation) ----
    zero_kernel<<<cdiv(N_NODES * D_H, TB), TB, 0, stream>>>(buf0, N_NODES * D_H);
    edge_fuse_kernel<<<cdiv(N_EDGES * 64, TB), TB, 0, stream>>>(
        buf2, ba, src, dst, buf0, N_EDGES);
    div_deg_kernel<<<cdiv(N_NODES * D_H, TB), TB, 0, stream>>>(buf0, deg, N_NODES, D_H);
    zero_kernel<<<cdiv(N_NODES * D_H, TB), TB, 0, stream>>>(buf1, N_NODES * D_H);
    gather_scatter_kernel<<<cdiv(N_EDGES * 64, TB), TB, 0, stream>>>(
        buf0, src, dst, buf1, N_EDGES, 64, D_H);
    // g2 = h2_l2 @ Wb^T
    wmma_gemm_nt_kernel<<<gemm_blocks, dim3(32, 8), 0, stream>>>(
        buf1, Wb, buf2, N_NODES, D_H, D_H);
    // mean over edges of lrelu(0.5*(g2[src]+g2[dst]) + bb)
    edge_reduce_kernel<<<cdiv(N_EDGES, 256), 256, 0, stream>>>(
        buf2, bb, src, dst, msum + br * D_H, N_EDGES);
  }

  final_kernel<<<1, 64, 0, stream>>>(msum, (const float*)d_in[14],
                                     (const float*)d_in[15], (float*)d_out);
}